// CEWithChunkedOutputLoss_79559974191144
// MI455X (gfx1250) — compile-verified
//
#include <hip/hip_runtime.h>
#include <hip/hip_bf16.h>

typedef __attribute__((ext_vector_type(2))) float v2f;
typedef __attribute__((ext_vector_type(8))) float v8f;
typedef __attribute__((ext_vector_type(4))) unsigned int v4u;

#define IGNORE_INDEX (-100LL)
#define L2E 1.4426950408889634f   // log2(e)
#define LN2 0.6931471805599453f

// Process 8 bf16 values (one b128 load) into running (max, sum) in log2 domain.
__device__ __forceinline__ void proc8(v4u q, float& m, float& s) {
    float x0 = __uint_as_float(q.x << 16)          * L2E;
    float x1 = __uint_as_float(q.x & 0xffff0000u)  * L2E;
    float x2 = __uint_as_float(q.y << 16)          * L2E;
    float x3 = __uint_as_float(q.y & 0xffff0000u)  * L2E;
    float x4 = __uint_as_float(q.z << 16)          * L2E;
    float x5 = __uint_as_float(q.z & 0xffff0000u)  * L2E;
    float x6 = __uint_as_float(q.w << 16)          * L2E;
    float x7 = __uint_as_float(q.w & 0xffff0000u)  * L2E;
    float cm = fmaxf(fmaxf(fmaxf(x0, x1), fmaxf(x2, x3)),
                     fmaxf(fmaxf(x4, x5), fmaxf(x6, x7)));
    if (cm > m) { s *= exp2f(m - cm); m = cm; }   // rare after warm-up
    float e0 = exp2f(x0 - m), e1 = exp2f(x1 - m);
    float e2 = exp2f(x2 - m), e3 = exp2f(x3 - m);
    float e4 = exp2f(x4 - m), e5 = exp2f(x5 - m);
    float e6 = exp2f(x6 - m), e7 = exp2f(x7 - m);
    s += ((e0 + e1) + (e2 + e3)) + ((e4 + e5) + (e6 + e7));
}

__device__ __forceinline__ void combine(float& m, float& s, float om, float os) {
    float nm = fmaxf(m, om);
    s = s * exp2f(m - nm) + os * exp2f(om - nm);
    m = nm;
}

// One block (256 threads = 8 wave32) per token row; online softmax over V.
__global__ void __launch_bounds__(256)
ce_row_kernel(const unsigned short* __restrict__ logits,
              const float* __restrict__ ref_lp,
              const long long* __restrict__ labels,
              float* __restrict__ contrib,
              float* __restrict__ validc,
              int V) {
    const int row = blockIdx.x;
    const int tid = threadIdx.x;
    const int nchunks = V >> 3;                       // 8 bf16 per b128
    const v4u* __restrict__ rp =
        (const v4u*)(logits + (size_t)row * (size_t)V);

    float m = -1.0e30f, s = 0.0f;

    // 4 batched NT b128 loads per iteration -> 4 outstanding loads / thread.
    for (int c = tid; c < nchunks; c += 4 * 256) {
        int c1 = c + 256, c2 = c + 512, c3 = c + 768;
        v4u q0 = __builtin_nontemporal_load(rp + c);
        v4u q1, q2, q3;
        bool h1 = c1 < nchunks, h2 = c2 < nchunks, h3 = c3 < nchunks;
        if (h1) q1 = __builtin_nontemporal_load(rp + c1);
        if (h2) q2 = __builtin_nontemporal_load(rp + c2);
        if (h3) q3 = __builtin_nontemporal_load(rp + c3);
        proc8(q0, m, s);
        if (h1) proc8(q1, m, s);
        if (h2) proc8(q2, m, s);
        if (h3) proc8(q3, m, s);
    }

    // wave32 reduction
    const int lane = tid & 31, wid = tid >> 5;
    #pragma unroll
    for (int off = 16; off >= 1; off >>= 1) {
        float om = __shfl_xor(m, off, 32);
        float os = __shfl_xor(s, off, 32);
        combine(m, s, om, os);
    }

    __shared__ float sm[8], ss[8];
    if (lane == 0) { sm[wid] = m; ss[wid] = s; }
    __syncthreads();

    if (wid == 0) {
        m = sm[lane & 7];
        s = ss[lane & 7];
        #pragma unroll
        for (int off = 4; off >= 1; off >>= 1) {   // offsets 4,2,1: reduce each 8-group
            float om = __shfl_xor(m, off, 32);
            float os = __shfl_xor(s, off, 32);
            combine(m, s, om, os);
        }
        if (lane == 0) {
            // lse in natural log: sum exp(x) = 2^(m + log2 s), m already in log2 domain
            float lse = LN2 * (m + log2f(s));
            long long lbl = labels[row];
            bool ign = (lbl == IGNORE_INDEX);
            long long cl = lbl < 0 ? 0 : (lbl >= (long long)V ? (long long)(V - 1) : lbl);
            unsigned int bits =
                (unsigned int)logits[(size_t)row * (size_t)V + (size_t)cl] << 16;
            float sel = __uint_as_float(bits) - lse;
            float loss = ign ? 0.0f : -sel;
            float ratio = ign ? 1.0f : exp2f((sel - ref_lp[row]) * L2E);
            ratio = fmaxf(ratio, 0.0f);
            contrib[row] = loss * ratio;
            validc[row] = ign ? 0.0f : 1.0f;
        }
    }
}

// Single wave32: reduce 2048 per-row partials with v_wmma_f32_16x16x4_f32
// (A = per-lane partials in the 16x4 striped layout, B = ones => D[m][n] = p_m + p_{m+16}).
__global__ void __launch_bounds__(32)
ce_reduce_kernel(const float* __restrict__ contrib,
                 const float* __restrict__ validc,
                 float* __restrict__ out, int n) {
    const int lane = threadIdx.x;
    float p = 0.0f, v = 0.0f;
    for (int i = lane; i < n; i += 32) {
        p += contrib[i];
        v += validc[i];
    }

    v2f b; b.x = 1.0f; b.y = 1.0f;                 // all-ones B (layout-invariant)
    v2f ap; ap.x = p; ap.y = 0.0f;                  // A[m][0]=p_m, A[m][2]=p_{m+16}
    v2f av; av.x = v; av.y = 0.0f;
    v8f cp = {}, cv = {};
    cp = __builtin_amdgcn_wmma_f32_16x16x4_f32(false, ap, false, b,
                                               (short)0, cp, false, false);
    cv = __builtin_amdgcn_wmma_f32_16x16x4_f32(false, av, false, b,
                                               (short)0, cv, false, false);

    float tp = ((cp[0] + cp[1]) + (cp[2] + cp[3])) + ((cp[4] + cp[5]) + (cp[6] + cp[7]));
    float tv = ((cv[0] + cv[1]) + (cv[2] + cv[3])) + ((cv[4] + cv[5]) + (cv[6] + cv[7]));
    tp += __shfl_xor(tp, 16, 32);                   // rows 0-7(+16-23) + rows 8-15(+24-31)
    tv += __shfl_xor(tv, 16, 32);

    if (lane == 0) out[0] = tp / tv;
}

extern "C" void kernel_launch(void* const* d_in, const int* in_sizes, int n_in,
                              void* d_out, int out_size, void* d_ws, size_t ws_size,
                              hipStream_t stream) {
    (void)n_in; (void)out_size; (void)ws_size;
    const unsigned short* logits = (const unsigned short*)d_in[0];  // bf16 bits
    const float* ref_lp = (const float*)d_in[1];
    const long long* labels = (const long long*)d_in[2];            // int64 per reference

    const int BT = in_sizes[1];              // B*T rows
    const int V = in_sizes[0] / BT;          // vocab

    float* contrib = (float*)d_ws;
    float* validc = contrib + BT;

    ce_row_kernel<<<BT, 256, 0, stream>>>(logits, ref_lp, labels, contrib, validc, V);
    ce_reduce_kernel<<<1, 32, 0, stream>>>(contrib, validc, (float*)d_out, BT);
}